// Attention_45732811767878
// MI455X (gfx1250) — compile-verified
//
#include <hip/hip_runtime.h>
#include <hip/hip_bf16.h>

typedef __attribute__((ext_vector_type(2))) float v2f;
typedef __attribute__((ext_vector_type(8))) float v8f;

#define NB   8
#define JX   1024
#define JQ   128
#define DD   512
#define TI   64           // i-rows per block (4 waves x 16 rows)
#define KS   128          // k-slab
#define PITCH 132         // LDS row pitch (floats): 16B-aligned rows, bank-stride 4

// ---------------------------------------------------------------------------
// Kernel 1: s = (h .* w_hu) @ u^T + r_i + c_j + b ; softmax_j ; u_a = P @ u
// writes: out[...,512:1024]=u_a, out[...,1024:1536]=h*u_a, mx[n,i]=max_j s
// ---------------------------------------------------------------------------
__global__ __launch_bounds__(128)
void attn_main_kernel(const float* __restrict__ h, const float* __restrict__ u,
                      const float* __restrict__ aw, const float* __restrict__ ab,
                      float* __restrict__ out, float* __restrict__ mx)
{
    __shared__ float A_lds[TI * PITCH];   // h.*w_hu slab, later P (probs)
    __shared__ float B_lds[JQ * PITCH];   // u k-slab, later transposed u d-chunk
    __shared__ float r_lds[TI];
    __shared__ float c_lds[JQ];

    const int tid  = threadIdx.x;
    const int wave = tid >> 5;
    const int lane = tid & 31;
    const int half = lane >> 4;
    const int l15  = lane & 15;

    const int n  = blockIdx.x >> 4;
    const int i0 = (blockIdx.x & 15) * TI;

    const float* hN = h + (size_t)n * JX * DD;
    const float* uN = u + (size_t)n * JQ * DD;

    // ---- prologue: c_j = u_j . w_u + b (all threads), r_i = h_i . w_h (half)
    {
        const float4* ur = (const float4*)(uN + (size_t)tid * DD);
        const float4* wr = (const float4*)(aw + DD);
        float acc = ab[0];
        #pragma unroll 4
        for (int k = 0; k < DD / 4; ++k) {
            float4 a = ur[k], b = wr[k];
            acc += a.x * b.x + a.y * b.y + a.z * b.z + a.w * b.w;
        }
        c_lds[tid] = acc;
    }
    if (tid < TI) {
        const float4* hr = (const float4*)(hN + (size_t)(i0 + tid) * DD);
        const float4* wr = (const float4*)(aw);
        float acc = 0.f;
        #pragma unroll 4
        for (int k = 0; k < DD / 4; ++k) {
            float4 a = hr[k], b = wr[k];
            acc += a.x * b.x + a.y * b.y + a.z * b.z + a.w * b.w;
        }
        r_lds[tid] = acc;
    }
    __syncthreads();

    // ---- phase 1: s-GEMM over K=512 in 4 slabs of 128
    v8f acc[8];
    #pragma unroll
    for (int jt = 0; jt < 8; ++jt) acc[jt] = (v8f){0.f,0.f,0.f,0.f,0.f,0.f,0.f,0.f};

    const int arow = (wave << 4) + l15;   // A-fragment row (M = lane%16)

    for (int s = 0; s < DD / KS; ++s) {
        const int k0 = s * KS;
        // stage A = h .* w_hu  (TI x KS)
        for (int t = tid; t < TI * (KS / 4); t += 128) {
            const int row = t >> 5, k = (t & 31) << 2;
            float4 hv = *(const float4*)(hN + (size_t)(i0 + row) * DD + k0 + k);
            float4 wv = *(const float4*)(aw + 2 * DD + k0 + k);
            float4 r; r.x = hv.x*wv.x; r.y = hv.y*wv.y; r.z = hv.z*wv.z; r.w = hv.w*wv.w;
            *(float4*)&A_lds[row * PITCH + k] = r;
        }
        // stage B = u slab (JQ x KS), row-major
        for (int t = tid; t < JQ * (KS / 4); t += 128) {
            const int row = t >> 5, k = (t & 31) << 2;
            *(float4*)&B_lds[row * PITCH + k] =
                *(const float4*)(uN + (size_t)row * DD + k0 + k);
        }
        __syncthreads();

        for (int kk = 0; kk < KS; kk += 4) {
            const int kb = kk + (half << 1);
            const float2 av = *(const float2*)&A_lds[arow * PITCH + kb];
            v2f a; a[0] = av.x; a[1] = av.y;
            #pragma unroll
            for (int jt = 0; jt < 8; ++jt) {
                const float2 bv = *(const float2*)&B_lds[(jt * 16 + l15) * PITCH + kb];
                v2f b; b[0] = bv.x; b[1] = bv.y;
                acc[jt] = __builtin_amdgcn_wmma_f32_16x16x4_f32(
                    false, a, false, b, (short)0, acc[jt], false, false);
            }
        }
        __syncthreads();
    }

    // ---- bias terms: row g lives at (wave*16 + half*8 + g), col = jt*16+l15
    float rv[8], cv[8];
    #pragma unroll
    for (int g = 0; g < 8; ++g) rv[g] = r_lds[wave * 16 + half * 8 + g];
    #pragma unroll
    for (int jt = 0; jt < 8; ++jt) cv[jt] = c_lds[jt * 16 + l15];
    #pragma unroll
    for (int jt = 0; jt < 8; ++jt)
        #pragma unroll
        for (int g = 0; g < 8; ++g)
            acc[jt][g] += rv[g] + cv[jt];

    // ---- softmax over j (row data: 16 lanes of one half x 8 jt VGPRs)
    #pragma unroll
    for (int g = 0; g < 8; ++g) {
        float m = acc[0][g];
        #pragma unroll
        for (int jt = 1; jt < 8; ++jt) m = fmaxf(m, acc[jt][g]);
        #pragma unroll
        for (int off = 1; off < 16; off <<= 1) m = fmaxf(m, __shfl_xor(m, off, 32));
        float ssum = 0.f;
        #pragma unroll
        for (int jt = 0; jt < 8; ++jt) {
            float p = __expf(acc[jt][g] - m);
            acc[jt][g] = p;
            ssum += p;
        }
        #pragma unroll
        for (int off = 1; off < 16; off <<= 1) ssum += __shfl_xor(ssum, off, 32);
        const float inv = __frcp_rn(ssum);
        #pragma unroll
        for (int jt = 0; jt < 8; ++jt) acc[jt][g] *= inv;
        if (l15 == 0)   // one lane per 16-half owns this row's max
            mx[(size_t)n * JX + i0 + wave * 16 + half * 8 + g] = m;
    }

    // ---- round-trip P through LDS to re-fragment as an A operand
    #pragma unroll
    for (int jt = 0; jt < 8; ++jt)
        #pragma unroll
        for (int g = 0; g < 8; ++g)
            A_lds[(wave * 16 + half * 8 + g) * PITCH + jt * 16 + l15] = acc[jt][g];
    __syncthreads();

    // ---- phase 2: u_a = P @ u  (K = JQ = 128), d in chunks of 128
    float* outN = out + ((size_t)n * JX + i0) * 2048;
    for (int dc = 0; dc < DD / 128; ++dc) {
        const int d0 = dc * 128;
        // stage u d-chunk TRANSPOSED: Bt[d][j] = u[j][d0+d] -> B frags are b64
        for (int t = tid; t < JQ * 32; t += 128) {
            const int j = t >> 5, k = (t & 31) << 2;
            const float4 uv = *(const float4*)(uN + (size_t)j * DD + d0 + k);
            B_lds[(k + 0) * PITCH + j] = uv.x;
            B_lds[(k + 1) * PITCH + j] = uv.y;
            B_lds[(k + 2) * PITCH + j] = uv.z;
            B_lds[(k + 3) * PITCH + j] = uv.w;
        }
        __syncthreads();

        v8f acc2[8];
        #pragma unroll
        for (int ct = 0; ct < 8; ++ct) acc2[ct] = (v8f){0.f,0.f,0.f,0.f,0.f,0.f,0.f,0.f};

        for (int kk = 0; kk < JQ; kk += 4) {
            const int kb = kk + (half << 1);
            const float2 av = *(const float2*)&A_lds[arow * PITCH + kb];
            v2f a; a[0] = av.x; a[1] = av.y;
            #pragma unroll
            for (int ct = 0; ct < 8; ++ct) {
                // B[k][d] = Bt[d][k]: contiguous pair at row d = ct*16+l15
                const float2 bv = *(const float2*)&B_lds[(ct * 16 + l15) * PITCH + kb];
                v2f b; b[0] = bv.x; b[1] = bv.y;
                acc2[ct] = __builtin_amdgcn_wmma_f32_16x16x4_f32(
                    false, a, false, b, (short)0, acc2[ct], false, false);
            }
        }

        // epilogue: u_a and h*u_a
        #pragma unroll
        for (int ct = 0; ct < 8; ++ct) {
            #pragma unroll
            for (int g = 0; g < 8; ++g) {
                const int row = wave * 16 + half * 8 + g;
                const int col = d0 + ct * 16 + l15;
                const float ua = acc2[ct][g];
                const float hv = hN[(size_t)(i0 + row) * DD + col];
                float* orow = outN + (size_t)row * 2048;
                orow[512 + col]  = ua;
                orow[1024 + col] = hv * ua;
            }
        }
        __syncthreads();
    }
}

// ---------------------------------------------------------------------------
// Kernel 2: a_h = softmax_i(mx[n,:]) ; h_a[n,d] = sum_i a_h[i] * h[n,i,d]
// grid: (NB * 4) blocks; each block owns a 128-wide d-slice of one batch
// ---------------------------------------------------------------------------
__global__ __launch_bounds__(256)
void attn_rowsoft_kernel(const float* __restrict__ h, const float* __restrict__ mx,
                         float* __restrict__ ha)
{
    __shared__ float red[256];
    __shared__ float a_lds[JX];
    const int n = blockIdx.x >> 2;
    const int dpart = blockIdx.x & 3;
    const int tid = threadIdx.x;
    const float* m = mx + (size_t)n * JX;

    float lm = -3.4e38f;
    for (int i = tid; i < JX; i += 256) lm = fmaxf(lm, m[i]);
    red[tid] = lm; __syncthreads();
    for (int s = 128; s > 0; s >>= 1) {
        if (tid < s) red[tid] = fmaxf(red[tid], red[tid + s]);
        __syncthreads();
    }
    const float M = red[0]; __syncthreads();

    float ls = 0.f;
    for (int i = tid; i < JX; i += 256) {
        const float e = __expf(m[i] - M);
        a_lds[i] = e;
        ls += e;
    }
    red[tid] = ls; __syncthreads();
    for (int s = 128; s > 0; s >>= 1) {
        if (tid < s) red[tid] += red[tid + s];
        __syncthreads();
    }
    const float inv = __frcp_rn(red[0]); __syncthreads();

    // partial dot: thread (ihalf, d) sums i in [ihalf*512, ihalf*512+512)
    const int d     = dpart * 128 + (tid & 127);
    const int ihalf = tid >> 7;
    const float* hp = h + (size_t)n * JX * DD + (size_t)ihalf * 512 * DD;
    float acc = 0.f;
    #pragma unroll 4
    for (int i = 0; i < 512; ++i) acc += a_lds[ihalf * 512 + i] * hp[(size_t)i * DD + d];
    red[tid] = acc; __syncthreads();
    if (tid < 128)
        ha[(size_t)n * DD + d] = (red[tid] + red[tid + 128]) * inv;
}

// ---------------------------------------------------------------------------
// Kernel 3: out[...,0:512] = h ; out[...,1536:2048] = h * h_a (broadcast)
// ---------------------------------------------------------------------------
__global__ __launch_bounds__(256)
void attn_tail_kernel(const float* __restrict__ h, const float* __restrict__ ha,
                      float* __restrict__ out)
{
    const int idx = blockIdx.x * 256 + threadIdx.x;     // float4 index
    const int n   = idx >> 17;                          // 1024*128 units per n
    const int rem = idx & ((JX * (DD / 4)) - 1);
    const int i   = rem >> 7;
    const int d4  = rem & 127;
    const float4 hv  = ((const float4*)h)[idx];
    const float4 hav = *(const float4*)(ha + (size_t)n * DD + d4 * 4);
    float* orow = out + ((size_t)n * JX + i) * 2048;
    ((float4*)orow)[d4] = hv;
    float4 p; p.x = hv.x*hav.x; p.y = hv.y*hav.y; p.z = hv.z*hav.z; p.w = hv.w*hav.w;
    *(float4*)(orow + 1536 + d4 * 4) = p;
}

extern "C" void kernel_launch(void* const* d_in, const int* in_sizes, int n_in,
                              void* d_out, int out_size, void* d_ws, size_t ws_size,
                              hipStream_t stream) {
    const float* h  = (const float*)d_in[0];
    const float* u  = (const float*)d_in[1];
    const float* aw = (const float*)d_in[2];
    const float* ab = (const float*)d_in[3];
    float* out = (float*)d_out;
    float* mx  = (float*)d_ws;                 // N*JX floats
    float* ha  = mx + (size_t)NB * JX;         // N*DD floats

    attn_main_kernel<<<dim3(NB * (JX / TI)), dim3(128), 0, stream>>>(h, u, aw, ab, out, mx);
    attn_rowsoft_kernel<<<dim3(NB * 4), dim3(256), 0, stream>>>(h, mx, ha);
    attn_tail_kernel<<<dim3((NB * JX * (DD / 4)) / 256), dim3(256), 0, stream>>>(h, ha, out);
}